// StepLoss_38242388804049
// MI455X (gfx1250) — compile-verified
//
#include <hip/hip_runtime.h>
#include <math.h>

// ---------------------------------------------------------------------------
// StepLoss: loss = mean_i( 1 - softmax(6*op[i,:])[target[i]] )
//   op: [B=131072, C=1000] f32, target: [B] int32, out: [1] f32
// HBM-bound: 524 MB single pass -> ~22.5 us floor at 23.3 TB/s.
// CDNA5 path: async global->LDS DMA (ASYNCcnt) with dual-offset immediates,
// LDS-resident two-pass softmax, ds gather of the target logit, wave32
// shuffle reductions, native v_exp_f32 (weight folded into base-2 scale).
// ---------------------------------------------------------------------------

static constexpr int   WAVE           = 32;     // gfx1250 wave32
static constexpr int   ROWS_PER_BLOCK = 8;      // 8 waves / block
static constexpr int   BLOCK          = WAVE * ROWS_PER_BLOCK;
static constexpr float K2 = 6.0f * 1.4426950408889634f; // WEIGHT * log2(e)

// Issue chunk I: 16B/lane async DMA. The 24-bit instruction offset is added to
// BOTH the global address (VADDR) and the LDS address (VDST), so one base
// address pair serves all chunks with compile-time offsets.
template <int I, int NITER, int NC4>
struct AsyncIssue {
  static __device__ __forceinline__ void go(const void* gbase, unsigned lbase, int lane) {
    if (lane + I * WAVE < NC4) {   // folds to 'true' for all but the tail chunk
      asm volatile("global_load_async_to_lds_b128 %0, %1, off offset:%2"
                   :: "v"(lbase), "v"(gbase), "n"(I * WAVE * 16)
                   : "memory");
    }
    AsyncIssue<I + 1, NITER, NC4>::go(gbase, lbase, lane);
  }
};
template <int NITER, int NC4>
struct AsyncIssue<NITER, NITER, NC4> {
  static __device__ __forceinline__ void go(const void*, unsigned, int) {}
};

__device__ __forceinline__ void wait_async0() {
  asm volatile("s_wait_asynccnt 0" ::: "memory");
}

// ---- specialized kernel: NC4 = C/4 known at compile time (C=1000 -> 250) ----
template <int NC4>
__global__ __launch_bounds__(BLOCK)
void rowloss_fixed_kernel(const float* __restrict__ op, const int* __restrict__ tgt,
                          float* __restrict__ partials, int B) {
  constexpr int NITER = (NC4 + WAVE - 1) / WAVE;   // 8 chunks for NC4=250
  constexpr int NPAD4 = NITER * WAVE;              // 256 float4 slots per row
  constexpr int C     = NC4 * 4;

  __shared__ float4 tile[ROWS_PER_BLOCK][NPAD4];   // 32 KB of 320 KB WGP LDS
  __shared__ float  wpart[ROWS_PER_BLOCK];

  const int lane = threadIdx.x & (WAVE - 1);
  const int wid  = threadIdx.x / WAVE;
  const int row  = blockIdx.x * ROWS_PER_BLOCK + wid;

  float contrib = 0.0f;
  if (row < B) {
    // Pad tail slots with -inf so compute passes need no predication
    // (max: no-op; v_exp_f32(-inf) = 0). Ordered before reads by the
    // "memory"-clobbered wait below + in-order per-wave DS.
    if (NPAD4 > NC4 && lane < NPAD4 - NC4)
      tile[wid][NC4 + lane] = make_float4(-INFINITY, -INFINITY, -INFINITY, -INFINITY);

    const float4* gbase = (const float4*)(op + (size_t)row * (size_t)C) + lane;
    const unsigned lbase = (unsigned)(unsigned long long)&tile[wid][lane];
    AsyncIssue<0, NITER, NC4>::go(gbase, lbase, lane);
    wait_async0();

    // ---- pass 1: row max (scale by K2 after; K2 > 0 so max commutes) ----
    float m = -INFINITY;
    #pragma unroll
    for (int i = 0; i < NITER; ++i) {
      float4 v = tile[wid][lane + i * WAVE];
      m = fmaxf(m, fmaxf(fmaxf(v.x, v.y), fmaxf(v.z, v.w)));
    }
    #pragma unroll
    for (int off = 16; off > 0; off >>= 1)
      m = fmaxf(m, __shfl_xor(m, off, WAVE));
    m *= K2;                                        // max of z = K2*x

    // ---- pass 2: sum of exp2(K2*x - m) ----
    float s = 0.0f;
    #pragma unroll
    for (int i = 0; i < NITER; ++i) {
      float4 v = tile[wid][lane + i * WAVE];
      s += __builtin_amdgcn_exp2f(fmaf(K2, v.x, -m));
      s += __builtin_amdgcn_exp2f(fmaf(K2, v.y, -m));
      s += __builtin_amdgcn_exp2f(fmaf(K2, v.z, -m));
      s += __builtin_amdgcn_exp2f(fmaf(K2, v.w, -m));
    }
    #pragma unroll
    for (int off = 16; off > 0; off >>= 1)
      s += __shfl_xor(s, off, WAVE);

    // ---- gather target logit from LDS, form 1 - p ----
    if (lane == 0) {
      const float* rowf = (const float*)&tile[wid][0];
      int t = tgt[row];
      contrib = 1.0f - __builtin_amdgcn_exp2f(fmaf(K2, rowf[t], -m)) / s;
    }
  }

  if (lane == 0) wpart[wid] = contrib;
  __syncthreads();
  if (threadIdx.x == 0) {
    float acc = 0.0f;
    #pragma unroll
    for (int i = 0; i < ROWS_PER_BLOCK; ++i) acc += wpart[i];
    partials[blockIdx.x] = acc;                     // deterministic per-block slot
  }
}

// ---- generic fallback for other C (dynamic loops, direct global loads) ----
__global__ __launch_bounds__(BLOCK)
void rowloss_dyn_kernel(const float* __restrict__ op, const int* __restrict__ tgt,
                        float* __restrict__ partials, int B, int C) {
  __shared__ float wpart[ROWS_PER_BLOCK];
  const int lane = threadIdx.x & (WAVE - 1);
  const int wid  = threadIdx.x / WAVE;
  const int row  = blockIdx.x * ROWS_PER_BLOCK + wid;

  float contrib = 0.0f;
  if (row < B) {
    const float* rowp = op + (size_t)row * (size_t)C;
    float m = -INFINITY;
    for (int j = lane; j < C; j += WAVE) m = fmaxf(m, rowp[j]);
    for (int off = 16; off > 0; off >>= 1) m = fmaxf(m, __shfl_xor(m, off, WAVE));
    m *= K2;
    float s = 0.0f;
    for (int j = lane; j < C; j += WAVE)
      s += __builtin_amdgcn_exp2f(fmaf(K2, rowp[j], -m));
    for (int off = 16; off > 0; off >>= 1) s += __shfl_xor(s, off, WAVE);
    if (lane == 0) {
      int t = tgt[row];
      contrib = 1.0f - __builtin_amdgcn_exp2f(fmaf(K2, rowp[t], -m)) / s;
    }
  }
  if (lane == 0) wpart[wid] = contrib;
  __syncthreads();
  if (threadIdx.x == 0) {
    float acc = 0.0f;
    #pragma unroll
    for (int i = 0; i < ROWS_PER_BLOCK; ++i) acc += wpart[i];
    partials[blockIdx.x] = acc;
  }
}

__global__ __launch_bounds__(256)
void final_reduce_kernel(const float* __restrict__ partials, int n,
                         float invB, float* __restrict__ out) {
  __shared__ float sdata[256];
  float acc = 0.0f;
  for (int i = threadIdx.x; i < n; i += 256)        // fixed order -> deterministic
    acc += partials[i];
  sdata[threadIdx.x] = acc;
  __syncthreads();
  #pragma unroll
  for (int s = 128; s > 0; s >>= 1) {
    if ((int)threadIdx.x < s) sdata[threadIdx.x] += sdata[threadIdx.x + s];
    __syncthreads();
  }
  if (threadIdx.x == 0) out[0] = sdata[0] * invB;
}

extern "C" void kernel_launch(void* const* d_in, const int* in_sizes, int n_in,
                              void* d_out, int out_size, void* d_ws, size_t ws_size,
                              hipStream_t stream) {
  (void)n_in; (void)out_size; (void)ws_size;
  const float* op  = (const float*)d_in[0];
  const int*   tgt = (const int*)d_in[1];
  const int B = in_sizes[1];                 // 131072
  const int C = in_sizes[0] / B;             // 1000
  float* partials = (float*)d_ws;            // one float per block (64 KB)

  const int nblocks = (B + ROWS_PER_BLOCK - 1) / ROWS_PER_BLOCK;  // 16384
  if (C == 1000) {
    rowloss_fixed_kernel<250><<<nblocks, BLOCK, 0, stream>>>(op, tgt, partials, B);
  } else {
    rowloss_dyn_kernel<<<nblocks, BLOCK, 0, stream>>>(op, tgt, partials, B, C);
  }
  final_reduce_kernel<<<1, 256, 0, stream>>>(partials, nblocks, 1.0f / (float)B,
                                             (float*)d_out);
}